// ProjectiveSampler_15822659518866
// MI455X (gfx1250) — compile-verified
//
#include <hip/hip_runtime.h>
#include <hip/hip_bf16.h>

typedef __attribute__((ext_vector_type(2))) float v2f;
typedef __attribute__((ext_vector_type(8))) float v8f;
typedef __attribute__((ext_vector_type(4))) unsigned int u32x4;
typedef __attribute__((ext_vector_type(4))) int i32x4;
typedef __attribute__((ext_vector_type(8))) int i32x8;

#define B_ 4
#define N_ 16384
#define T_ 16
#define C_ 128
#define H_ 64
#define W_ 64
#define O_ 256

// ---------------------------------------------------------------------------
// Kernel 1: (B,T,C,H,W) -> (B,T,H,W,C) tiled transpose (per (b,t): 128 x 4096)
// ---------------------------------------------------------------------------
__global__ __launch_bounds__(256) void k_transpose(const float* __restrict__ in,
                                                   float* __restrict__ out) {
  __shared__ float tile[32][33];
  const int bt  = blockIdx.y;        // 0..63  (b*T+t)
  const int ct  = blockIdx.x & 3;    // channel tile 0..3
  const int hwt = blockIdx.x >> 2;   // hw tile 0..127
  const int tx  = threadIdx.x & 31;
  const int ty  = threadIdx.x >> 5;  // 0..7
  const int c0  = ct * 32;
  const int hw0 = hwt * 32;
  const float* src = in  + (size_t)bt * C_ * (H_ * W_);
  float*       dst = out + (size_t)bt * (H_ * W_) * C_;
#pragma unroll
  for (int i = 0; i < 4; ++i) {
    int r = ty + 8 * i;
    tile[r][tx] = src[(size_t)(c0 + r) * (H_ * W_) + hw0 + tx];
  }
  __syncthreads();
#pragma unroll
  for (int i = 0; i < 4; ++i) {
    int r = ty + 8 * i;
    dst[(size_t)(hw0 + r) * C_ + c0 + tx] = tile[tx][r];
  }
}

// ---------------------------------------------------------------------------
// Kernel 2: project + bilinear gather (coalesced 512B bursts) + temporal mean
// One wave per point; each lane owns 4 channels (float4).
// ---------------------------------------------------------------------------
__device__ __forceinline__ void corner_acc(const float* __restrict__ base,
                                           int yi, int xi, float w, int lane,
                                           float4& acc) {
  bool valid = (xi >= 0) & (xi < W_) & (yi >= 0) & (yi < H_);
  int xc = xi < 0 ? 0 : (xi > W_ - 1 ? W_ - 1 : xi);
  int yc = yi < 0 ? 0 : (yi > H_ - 1 ? H_ - 1 : yi);
  const float4* p = (const float4*)(base + (size_t)(yc * W_ + xc) * C_) + lane;
  float4 v = *p;
  float wv = valid ? w : 0.0f;
  acc.x += wv * v.x; acc.y += wv * v.y; acc.z += wv * v.z; acc.w += wv * v.w;
}

__global__ __launch_bounds__(256) void k_project_gather(
    const float* __restrict__ pts, const float* __restrict__ featT,
    const float* __restrict__ Kmat, const float* __restrict__ ext,
    float* __restrict__ xmean) {
  const int wave = threadIdx.x >> 5;
  const int lane = threadIdx.x & 31;
  const int pidx = blockIdx.x * 8 + wave;  // b*N + n
  const int b = pidx >> 14;                // N_ == 16384

  const float px = pts[pidx * 3 + 0];
  const float py = pts[pidx * 3 + 1];
  const float pz = pts[pidx * 3 + 2];

  const float* Kb = Kmat + b * 9;
  const float K00 = Kb[0], K01 = Kb[1], K02 = Kb[2];
  const float K10 = Kb[3], K11 = Kb[4], K12 = Kb[5];

  float4 acc = {0.f, 0.f, 0.f, 0.f};
  for (int t = 0; t < T_; ++t) {
    const float* E = ext + (size_t)(b * T_ + t) * 16;
    float pc0 = E[0] * px + E[1] * py + E[2]  * pz + E[3];
    float pc1 = E[4] * px + E[5] * py + E[6]  * pz + E[7];
    float pc2 = E[8] * px + E[9] * py + E[10] * pz + E[11];
    float z   = fmaxf(pc2, 1e-6f);
    float inv = 1.0f / z;
    float xh = pc0 * inv, yh = pc1 * inv, zh = pc2 * inv;
    float u = K00 * xh + K01 * yh + K02 * zh;
    float v = K10 * xh + K11 * yh + K12 * zh;
    // x = u*(W-1)/W ; y = v*(H-1)/H  (algebraic reduction of grid mapping)
    float x = u * ((float)(W_ - 1) / (float)W_);
    float y = v * ((float)(H_ - 1) / (float)H_);
    float xf = floorf(x), yf = floorf(y);
    int x0 = (int)xf, y0 = (int)yf;
    int x1 = x0 + 1,  y1 = y0 + 1;
    float wx1 = x - xf, wx0 = 1.0f - wx1;
    float wy1 = y - yf, wy0 = 1.0f - wy1;

    const float* base = featT + (size_t)(b * T_ + t) * (H_ * W_) * C_;
    corner_acc(base, y0, x0, wy0 * wx0, lane, acc);
    corner_acc(base, y0, x1, wy0 * wx1, lane, acc);
    corner_acc(base, y1, x0, wy1 * wx0, lane, acc);
    corner_acc(base, y1, x1, wy1 * wx1, lane, acc);
  }
  const float s = 1.0f / (float)T_;
  float4 o = {acc.x * s, acc.y * s, acc.z * s, acc.w * s};
  *(((float4*)(xmean + (size_t)pidx * C_)) + lane) = o;
}

// ---------------------------------------------------------------------------
// TDM: load a 32x128 f32 tile into LDS with row padding 128dw + 4dw = 132
// (pad_interval=6 -> every 128 DWORDs, pad_amount=3 -> 4 DWORDs) so the LDS
// image directly matches the bank-conflict-free stride LDX=132.
// ---------------------------------------------------------------------------
#define LDX 132
#define LDH 260

__device__ __forceinline__ void tdm_load_x_tile(const float* gsrc,
                                                void* lds_dst) {
  unsigned long long ga = (unsigned long long)(uintptr_t)gsrc;
  unsigned lds = (unsigned)(uintptr_t)lds_dst;  // low 32 bits = LDS byte addr
  u32x4 g0;
  g0[0] = 1u;                                   // count=1, user descriptor
  g0[1] = lds;                                  // lds_addr
  g0[2] = (unsigned)(ga & 0xFFFFFFFFu);         // global_addr[31:0]
  g0[3] = (unsigned)((ga >> 32) & 0x01FFFFFFu)  // global_addr[56:32]
          | (2u << 30);                         // type = 2 ("image")
  i32x8 g1;
  g1[0] = (2 << 16)      // data_size = 4 bytes
        | (1 << 20)      // pad_enable
        | (6 << 22)      // pad_interval: every 128 DWORDs
        | (3 << 25);     // pad_amount: 4 DWORDs
  g1[1] = (int)(128u << 16);   // tensor_dim0 = 128
  g1[2] = (int)(32u << 16);    // tensor_dim0 hi | tensor_dim1 = 32
  g1[3] = (int)(128u << 16);   // tensor_dim1 hi | tile_dim0 = 128
  g1[4] = 32;                  // tile_dim1 = 32, tile_dim2 = 0
  g1[5] = 128;                 // tensor_dim0_stride = 128
  g1[6] = 0;
  g1[7] = 0;
  i32x4 gz = {0, 0, 0, 0};
#if __clang_major__ >= 23
  i32x8 gz8 = {};
  __builtin_amdgcn_tensor_load_to_lds(g0, g1, gz, gz, gz8, 0);
#else
  __builtin_amdgcn_tensor_load_to_lds(g0, g1, gz, gz, 0);
#endif
}

// ---------------------------------------------------------------------------
// Kernel 3: fused MLP with V_WMMA_F32_16X16X4_F32.
// Block = 256 thr = 8 waves, 32 rows x 256 cols. X tile staged via TDM.
// Each wave: m-tile = wave&1, four n-tiles processed as 2 pairs with two
// independent accumulator chains sharing one A-fragment.
//   Stage 1: H = relu(X @ W1^T + b1)  (K=128)  -> LDS
//   Stage 2: out = H @ W2^T + b2      (K=256)
// A-frag (16x4 f32): lane l holds A[l&15][2*(l>>4)+{0,1}]
// B-frag (4x16 f32): lane l holds B[2*(l>>4)+{0,1}][l&15]  (B[k][n] = W[n][k])
// C/D: vgpr r -> row r+8*(l>>4), col l&15
// ---------------------------------------------------------------------------
__global__ __launch_bounds__(256) void k_mlp_wmma(
    const float* __restrict__ X, const float* __restrict__ Wh,
    const float* __restrict__ bh, const float* __restrict__ Wo,
    const float* __restrict__ bo, float* __restrict__ out) {
  __shared__ float Xlds[32 * LDX];
  __shared__ float Hlds[32 * LDH];
  const int tid  = threadIdx.x;
  const int wave = tid >> 5;
  const int lane = tid & 31;
  const int l16  = lane & 15;
  const int half = lane >> 4;
  const int rowBase = blockIdx.x * 32;

  // Stage X tile (32x128) into padded LDS via the Tensor Data Mover.
  if (wave == 0) {
    tdm_load_x_tile(X + (size_t)rowBase * C_, (void*)Xlds);
    __builtin_amdgcn_s_wait_tensorcnt(0);
  }
  __syncthreads();

  const int m0  = (wave & 1) * 16;
  const int ntb = wave >> 1;           // 0..3 -> n-tiles ntb*4 .. ntb*4+3

  // ---- Stage 1: K = 128 ----
  for (int p = 0; p < 2; ++p) {
    const int nA = (ntb * 4 + 2 * p) * 16;
    const int nB = nA + 16;
    v8f acc0 = {}, acc1 = {};
    const float* arow = &Xlds[(m0 + l16) * LDX + 2 * half];
    const float* bp0  = Wh + (size_t)(nA + l16) * C_ + 2 * half;
    const float* bp1  = Wh + (size_t)(nB + l16) * C_ + 2 * half;
#pragma unroll 4
    for (int k0 = 0; k0 < C_; k0 += 4) {
      v2f a;  a[0]  = arow[k0]; a[1]  = arow[k0 + 1];
      v2f f0; f0[0] = bp0[k0];  f0[1] = bp0[k0 + 1];
      v2f f1; f1[0] = bp1[k0];  f1[1] = bp1[k0 + 1];
      acc0 = __builtin_amdgcn_wmma_f32_16x16x4_f32(false, a, false, f0,
                                                   (short)0, acc0, false, false);
      acc1 = __builtin_amdgcn_wmma_f32_16x16x4_f32(false, a, false, f1,
                                                   (short)0, acc1, false, false);
    }
    const float biasA = bh[nA + l16];
    const float biasB = bh[nB + l16];
#pragma unroll
    for (int r = 0; r < 8; ++r) {
      float h0 = acc0[r] + biasA; h0 = h0 > 0.0f ? h0 : 0.0f;
      float h1 = acc1[r] + biasB; h1 = h1 > 0.0f ? h1 : 0.0f;
      Hlds[(m0 + r + 8 * half) * LDH + nA + l16] = h0;
      Hlds[(m0 + r + 8 * half) * LDH + nB + l16] = h1;
    }
  }
  __syncthreads();

  // ---- Stage 2: K = 256 ----
  for (int p = 0; p < 2; ++p) {
    const int nA = (ntb * 4 + 2 * p) * 16;
    const int nB = nA + 16;
    v8f acc0 = {}, acc1 = {};
    const float* arow = &Hlds[(m0 + l16) * LDH + 2 * half];
    const float* bp0  = Wo + (size_t)(nA + l16) * O_ + 2 * half;
    const float* bp1  = Wo + (size_t)(nB + l16) * O_ + 2 * half;
#pragma unroll 4
    for (int k0 = 0; k0 < O_; k0 += 4) {
      v2f a;  a[0]  = arow[k0]; a[1]  = arow[k0 + 1];
      v2f f0; f0[0] = bp0[k0];  f0[1] = bp0[k0 + 1];
      v2f f1; f1[0] = bp1[k0];  f1[1] = bp1[k0 + 1];
      acc0 = __builtin_amdgcn_wmma_f32_16x16x4_f32(false, a, false, f0,
                                                   (short)0, acc0, false, false);
      acc1 = __builtin_amdgcn_wmma_f32_16x16x4_f32(false, a, false, f1,
                                                   (short)0, acc1, false, false);
    }
    const float biasA = bo[nA + l16];
    const float biasB = bo[nB + l16];
#pragma unroll
    for (int r = 0; r < 8; ++r) {
      out[(size_t)(rowBase + m0 + r + 8 * half) * O_ + nA + l16] =
          acc0[r] + biasA;
      out[(size_t)(rowBase + m0 + r + 8 * half) * O_ + nB + l16] =
          acc1[r] + biasB;
    }
  }
}

// ---------------------------------------------------------------------------
extern "C" void kernel_launch(void* const* d_in, const int* in_sizes, int n_in,
                              void* d_out, int out_size, void* d_ws, size_t ws_size,
                              hipStream_t stream) {
  (void)in_sizes; (void)n_in; (void)out_size; (void)ws_size;
  const float* pts  = (const float*)d_in[0];
  const float* feat = (const float*)d_in[1];
  const float* Kmat = (const float*)d_in[2];
  const float* ext  = (const float*)d_in[3];
  const float* w1   = (const float*)d_in[4];
  const float* b1   = (const float*)d_in[5];
  const float* w2   = (const float*)d_in[6];
  const float* b2   = (const float*)d_in[7];
  float* out = (float*)d_out;

  float* featT = (float*)d_ws;                                   // 134.2 MB
  float* xmean = featT + (size_t)B_ * T_ * H_ * W_ * C_;         // + 33.5 MB

  dim3 gA(512, B_ * T_);
  k_transpose<<<gA, 256, 0, stream>>>(feat, featT);

  k_project_gather<<<(B_ * N_) / 8, 256, 0, stream>>>(pts, featT, Kmat, ext,
                                                      xmean);

  k_mlp_wmma<<<(B_ * N_) / 32, 256, 0, stream>>>(xmean, w1, b1, w2, b2, out);
}